// DepthSSIMLoss_74483322847627
// MI455X (gfx1250) — compile-verified
//
#include <hip/hip_runtime.h>
#include <math.h>

typedef __attribute__((ext_vector_type(2))) float v2f;
typedef __attribute__((ext_vector_type(8))) float v8f;

#define IMG_H 512
#define IMG_W 512
#define NB 32
#define NPIX (NB * IMG_H * IMG_W) /* 8388608 per tensor */
#define NTILES (NB * 32 * 32)     /* 32768 16x16 tiles  */
#define WAVES_PER_BLK 4
#define BLK_THREADS (WAVES_PER_BLK * 32)
#define NBLK_TILE (NTILES / WAVES_PER_BLK) /* 8192 blocks */

/* per-wave LDS layout in floats: 3 padded patches (32x28) + Hq (32x16) */
#define PM_OFF 0
#define PIM_OFF 896
#define PTM_OFF 1792
#define HQ_OFF 2688
#define WAVE_LDS 3200
#define G_OFF (WAVE_LDS * WAVES_PER_BLK)   /* 12800 */
#define RED_OFF (G_OFF + 16)               /* 12816 */
#define LDS_TOTAL (RED_OFF + BLK_THREADS)  /* 12944 floats = 51.8 KB */

/* order-preserving float<->uint for deterministic atomic max */
__device__ __forceinline__ unsigned enc_f(float f) {
  unsigned u = __float_as_uint(f);
  return (u & 0x80000000u) ? ~u : (u | 0x80000000u);
}
__device__ __forceinline__ float dec_f(unsigned k) {
  unsigned b = (k & 0x80000000u) ? (k ^ 0x80000000u) : ~k;
  return __uint_as_float(b);
}

/* recover separable 1-D gaussian g[i] = sum_j w2d[i][j] (since sum(g)==1) */
__global__ void k_init(const float* __restrict__ win, unsigned* wsu, float* wsf) {
  int t = threadIdx.x;
  if (t == 0) wsu[0] = 0u;
  if (t < 11) {
    float s = 0.f;
    for (int j = 0; j < 11; ++j) s += win[t * 11 + j];
    wsf[2 + t] = s;
  }
}

__global__ void k_max(const float* __restrict__ in, const float* __restrict__ tg,
                      const float* __restrict__ mk, unsigned* wsu) {
  __shared__ unsigned smx[256];
  unsigned best = 0u;
  long long stride = (long long)gridDim.x * blockDim.x;
  for (long long i = (long long)blockIdx.x * blockDim.x + threadIdx.x; i < NPIX; i += stride) {
    float m = mk[i];
    unsigned a = enc_f(in[i] * m);
    unsigned b = enc_f(tg[i] * m);
    best = max(best, max(a, b));
  }
  smx[threadIdx.x] = best;
  __syncthreads();
  for (int o = 128; o > 0; o >>= 1) {
    if (threadIdx.x < (unsigned)o) smx[threadIdx.x] = max(smx[threadIdx.x], smx[threadIdx.x + o]);
    __syncthreads();
  }
  if (threadIdx.x == 0) atomicMax(wsu, smx[0]);
}

__device__ __forceinline__ float patchval(int q, const float* pm, const float* pim,
                                          const float* ptm, int row, int col) {
  int o = row * 28 + col;
  switch (q) {
    case 0: return pm[o];
    case 1: return pim[o];
    case 2: return ptm[o];
    case 3: { float v = pim[o]; return v * v; }
    case 4: { float v = ptm[o]; return v * v; }
    default: return pim[o] * ptm[o];
  }
}

__global__ __launch_bounds__(BLK_THREADS) void k_ssim(
    const float* __restrict__ IN, const float* __restrict__ TG, const float* __restrict__ MK,
    const unsigned* __restrict__ wsu, const float* __restrict__ wsf, float* __restrict__ partials) {
  __shared__ float lds[LDS_TOTAL];
  const int tid = threadIdx.x;
  const int lane = tid & 31, wv = tid >> 5;
  const int l16 = lane & 15, half = lane >> 4;

  float* pw = lds + wv * WAVE_LDS;
  float* pm = pw + PM_OFF;
  float* pim = pw + PIM_OFF;
  float* ptm = pw + PTM_OFF;
  float* hq = pw + HQ_OFF;
  float* sg = lds + G_OFF;

  if (tid < 11) sg[tid] = wsf[2 + tid];
  __syncthreads();

  int tile = blockIdx.x * WAVES_PER_BLK + wv;
  int b = tile >> 10;
  int rem = tile & 1023;
  int ty = rem >> 5, tx = rem & 31;
  const float* I0 = IN + (size_t)b * (IMG_H * IMG_W);
  const float* T0 = TG + (size_t)b * (IMG_H * IMG_W);
  const float* M0 = MK + (size_t)b * (IMG_H * IMG_W);
  int r0 = ty * 16, c0 = tx * 16;

  /* load 26x26 halo patch (zero-padded to 32x28): m, i*m, t*m */
  for (int it = 0; it < 28; ++it) {
    int idx = it * 32 + lane;
    int row = idx / 28, col = idx - row * 28;
    float mv = 0.f, iv = 0.f, tv = 0.f;
    int gr = r0 - 5 + row, gc = c0 - 5 + col;
    if (row < 26 && col < 26 && (unsigned)gr < IMG_H && (unsigned)gc < IMG_W) {
      int off = gr * IMG_W + gc;
      mv = M0[off];
      iv = I0[off] * mv;
      tv = T0[off] * mv;
    }
    pm[row * 28 + col] = mv;
    pim[row * 28 + col] = iv;
    ptm[row * 28 + col] = tv;
  }
  __syncthreads();

  /* Banded-gaussian operand registers. Under the wave32 f32 layouts the
     horizontal B-matrix Gh[k][n]=g[k-n] and vertical A-matrix Gv[m][k]=g[k-m]
     map to the SAME per-lane values: vgpr j holds G(K=4s+j+2*half, l16). */
  v2f wk[7];
#pragma unroll
  for (int s = 0; s < 7; ++s) {
    int k0 = 4 * s + 2 * half;
    int d0 = k0 - l16, d1 = d0 + 1;
    wk[s].x = ((unsigned)d0 <= 10u) ? sg[d0] : 0.f;
    wk[s].y = ((unsigned)d1 <= 10u) ? sg[d1] : 0.f;
  }

  v8f out[6];
#pragma unroll
  for (int q = 0; q < 6; ++q) {
    /* horizontal pass: Hq(32x16) = P(32x28) x Gh(28x16), two M blocks */
#pragma unroll
    for (int mb = 0; mb < 2; ++mb) {
      v8f acc = {};
      int row = mb * 16 + l16;
#pragma unroll
      for (int s = 0; s < 7; ++s) {
        int col = 4 * s + 2 * half;
        v2f a;
        a.x = patchval(q, pm, pim, ptm, row, col);
        a.y = patchval(q, pm, pim, ptm, row, col + 1);
        acc = __builtin_amdgcn_wmma_f32_16x16x4_f32(false, a, false, wk[s],
                                                    (short)0, acc, false, false);
      }
#pragma unroll
      for (int j = 0; j < 8; ++j) {
        int rr = mb * 16 + j + 8 * half; /* C/D layout row */
        hq[rr * 16 + l16] = acc[j];
      }
    }
    __syncthreads();
    /* vertical pass: Out(16x16) = Gv(16x28) x Hq(28x16) */
    v8f acc = {};
#pragma unroll
    for (int s = 0; s < 7; ++s) {
      int k0 = 4 * s + 2 * half;
      v2f bb;
      bb.x = hq[k0 * 16 + l16];
      bb.y = hq[(k0 + 1) * 16 + l16];
      acc = __builtin_amdgcn_wmma_f32_16x16x4_f32(false, wk[s], false, bb,
                                                  (short)0, acc, false, false);
    }
    out[q] = acc;
    __syncthreads(); /* WAR before next quantity overwrites hq */
  }

  /* elementwise structure map; all six outs share the C/D register layout */
  float L = dec_f(wsu[0]);
  float C2 = 0.03f * L;
  C2 = C2 * C2;
  float C3 = 0.5f * C2;
  float lsum = 0.f;
#pragma unroll
  for (int j = 0; j < 8; ++j) {
    float bm = out[0][j], bi = out[1][j], bt = out[2][j];
    float bii = out[3][j], btt = out[4][j], bit = out[5][j];
    float mw = bm + 1e-8f;
    float mu_i = bi / mw, mu_t = bt / mw;
    float sii = bii / mw - mu_i * mu_i;
    float stt = btt / mw - mu_t * mu_t;
    float sit = bit / mw - mu_i * mu_t;
    float si = sqrtf(fmaxf(sii, 0.f) + 1e-12f);
    float st = sqrtf(fmaxf(stt, 0.f) + 1e-12f);
    lsum += (sit + C3) / (si * st + C3 + 1e-8f);
  }

  float* red = lds + RED_OFF;
  red[tid] = lsum;
  __syncthreads();
  for (int o = BLK_THREADS / 2; o > 0; o >>= 1) {
    if (tid < o) red[tid] += red[tid + o];
    __syncthreads();
  }
  if (tid == 0) partials[blockIdx.x] = red[0];
}

__global__ void k_final(const float* __restrict__ partials, float* __restrict__ out) {
  __shared__ double sd[256];
  double s = 0.0;
  for (int i = threadIdx.x; i < NBLK_TILE; i += 256) s += (double)partials[i];
  sd[threadIdx.x] = s;
  __syncthreads();
  for (int o = 128; o > 0; o >>= 1) {
    if (threadIdx.x < o) sd[threadIdx.x] += sd[threadIdx.x + o];
    __syncthreads();
  }
  if (threadIdx.x == 0) out[0] = (float)(1.0 - sd[0] / (double)NPIX);
}

extern "C" void kernel_launch(void* const* d_in, const int* in_sizes, int n_in,
                              void* d_out, int out_size, void* d_ws, size_t ws_size,
                              hipStream_t stream) {
  const float* IN = (const float*)d_in[0];
  const float* TG = (const float*)d_in[1];
  const float* MK = (const float*)d_in[2];
  const float* WIN = (const float*)d_in[3];
  unsigned* wsu = (unsigned*)d_ws;
  float* wsf = (float*)d_ws;
  float* partials = wsf + 16;

  k_init<<<1, 64, 0, stream>>>(WIN, wsu, wsf);
  k_max<<<2048, 256, 0, stream>>>(IN, TG, MK, wsu);
  k_ssim<<<NBLK_TILE, BLK_THREADS, 0, stream>>>(IN, TG, MK, wsu, wsf, partials);
  k_final<<<1, 256, 0, stream>>>(partials, (float*)d_out);
}